// GCN_35802847380158
// MI455X (gfx1250) — compile-verified
//
#include <hip/hip_runtime.h>

#define NUM_NODES 100000
#define NUM_EDGES 1200000
#define NFEAT     128
#define NHID      64
#define NCLASS    16
#define NLAYER    4
#define ALPHA_F   0.1f

typedef __attribute__((ext_vector_type(2))) float v2f;
typedef __attribute__((ext_vector_type(8))) float v8f;

// ---------------------------------------------------------------------------
// Generic fp32 WMMA GEMM:  C[nrows,ncols] = A[nrows,K] @ B[K,ncols] (+ bias)
// Each wave computes one 16x16 output tile via V_WMMA_F32_16X16X4_F32,
// stepping K by 4. nrows must be a multiple of 16 (100000 = 6250*16),
// ncols a multiple of 16 (64 or 16), K a multiple of 4 (128 or 64).
// out      : raw result (+bias)          (nullable)
// out_relu : relu(result + bias)         (nullable)
// ---------------------------------------------------------------------------
__global__ __launch_bounds__(128) void gcn_wmma_gemm_f32(
    const float* __restrict__ A, const float* __restrict__ B,
    const float* __restrict__ bias,
    float* __restrict__ out, float* __restrict__ out_relu,
    int nrows, int K, int ncols)
{
    const int lane  = threadIdx.x & 31;
    const int wave  = threadIdx.x >> 5;
    const int m_tile = blockIdx.x * 4 + wave;          // wave-uniform
    if (m_tile * 16 >= nrows) return;                  // uniform exit: EXEC stays full
    const int m0 = m_tile * 16;
    const int n0 = blockIdx.y * 16;

    const int half = lane >> 4;        // 0: lanes 0-15, 1: lanes 16-31
    const int l15  = lane & 15;
    const int kb   = half * 2;         // A/B K-pair offset per ISA VGPR layout

    const float* arow = A + (size_t)(m0 + l15) * (size_t)K;  // A: M = l15
    const int    bn   = n0 + l15;                            // B: N = l15

    v8f acc = {};
    for (int k = 0; k < K; k += 4) {
        // A 16x4 fragment: v0 = A[m, k+kb], v1 = A[m, k+kb+1]
        v2f a = *(const v2f*)(arow + k + kb);
        // B 4x16 fragment (mirrored layout): v0 = B[k+kb, n], v1 = B[k+kb+1, n]
        v2f b;
        b.x = B[(size_t)(k + kb)     * (size_t)ncols + bn];
        b.y = B[(size_t)(k + kb + 1) * (size_t)ncols + bn];
        acc = __builtin_amdgcn_wmma_f32_16x16x4_f32(
                  /*neg_a=*/false, a, /*neg_b=*/false, b,
                  /*c_mod=*/(short)0, acc, /*reuse_a=*/false, /*reuse_b=*/false);
    }

    const float bv = bias ? bias[bn] : 0.0f;
    #pragma unroll
    for (int j = 0; j < 8; ++j) {
        // C/D layout: VGPR j -> row m0 + j + 8*half, col n0 + l15
        const int    row = m0 + j + half * 8;
        const size_t idx = (size_t)row * (size_t)ncols + (size_t)bn;
        const float  v   = acc[j] + bv;
        if (out)      out[idx]      = v;
        if (out_relu) out_relu[idx] = fmaxf(v, 0.0f);
    }
}

// ---------------------------------------------------------------------------
// acc = 0
// ---------------------------------------------------------------------------
__global__ void gcn_zero_f32(float* __restrict__ p, int n)
{
    int i = blockIdx.x * blockDim.x + threadIdx.x;
    if (i < n) p[i] = 0.0f;
}

// ---------------------------------------------------------------------------
// SpMM scatter: one wave per edge; 64 feature floats = 2 per lane.
// acc[rows[e], :] += vals[e] * xw[cols[e], :]
// Edge scalars are wave-uniform -> scalar loads; feature rows live in L2.
// ---------------------------------------------------------------------------
__global__ __launch_bounds__(256) void gcn_spmm_scatter(
    const int* __restrict__ rows, const int* __restrict__ cols,
    const float* __restrict__ vals,
    const float* __restrict__ xw, float* __restrict__ acc, int nedges)
{
    const int e    = (int)((blockIdx.x * blockDim.x + threadIdx.x) >> 5);
    const int lane = threadIdx.x & 31;
    if (e >= nedges) return;
    const int   r = rows[e];
    const int   c = cols[e];
    const float v = vals[e];
    const float* __restrict__ src = xw  + (size_t)c * NHID;
    float*       __restrict__ dst = acc + (size_t)r * NHID;
    atomicAdd(&dst[lane],      v * src[lane]);
    atomicAdd(&dst[lane + 32], v * src[lane + 32]);
}

// ---------------------------------------------------------------------------
// li = relu((1-alpha)*acc + alpha*h0 + b[col])   (theta term cancels: out = support + b)
// ---------------------------------------------------------------------------
__global__ void gcn_combine_relu(
    const float* __restrict__ acc, const float* __restrict__ h0,
    const float* __restrict__ bias, float* __restrict__ li, int total)
{
    int i = blockIdx.x * blockDim.x + threadIdx.x;
    if (i >= total) return;
    const int col = i & (NHID - 1);
    const float v = (1.0f - ALPHA_F) * acc[i] + ALPHA_F * h0[i] + bias[col];
    li[i] = fmaxf(v, 0.0f);
}

// ---------------------------------------------------------------------------
// Launch pipeline
// ---------------------------------------------------------------------------
extern "C" void kernel_launch(void* const* d_in, const int* in_sizes, int n_in,
                              void* d_out, int out_size, void* d_ws, size_t ws_size,
                              hipStream_t stream)
{
    const float* x      = (const float*)d_in[0];
    const int*   rows   = (const int*)  d_in[1];
    const int*   cols   = (const int*)  d_in[2];
    const float* vals   = (const float*)d_in[3];
    const float* fc0_w  = (const float*)d_in[4];
    const float* fc0_b  = (const float*)d_in[5];
    const float* conv_w = (const float*)d_in[6];   // [NLAYER, 64, 64]
    const float* conv_b = (const float*)d_in[7];   // [NLAYER, 64]
    const float* fc1_w  = (const float*)d_in[8];
    const float* fc1_b  = (const float*)d_in[9];
    float*       out    = (float*)d_out;

    const size_t NH = (size_t)NUM_NODES * NHID;    // 6.4M floats per buffer
    float* h0  = (float*)d_ws;
    float* li  = h0 + NH;
    float* xw  = li + NH;
    float* acc = xw + NH;

    const int m_tiles      = NUM_NODES / 16;               // 6250 (exact)
    const dim3 gemm_blk(128);                              // 4 waves / block
    const dim3 gemm_grid_hid((m_tiles + 3) / 4, NHID / 16);   // x 4 col-tiles
    const dim3 gemm_grid_cls((m_tiles + 3) / 4, NCLASS / 16); // x 1 col-tile

    const int elem_total   = (int)NH;
    const int elem_blocks  = (elem_total + 255) / 256;
    const int edge_blocks  = (NUM_EDGES + 7) / 8;          // 8 waves / block

    // Stage 1: h0 = x @ fc0_w + fc0_b ; li = relu(h0)
    gcn_wmma_gemm_f32<<<gemm_grid_hid, gemm_blk, 0, stream>>>(
        x, fc0_w, fc0_b, h0, li, NUM_NODES, NFEAT, NHID);

    // Stage 2: 4 GCNII layers
    for (int i = 0; i < NLAYER; ++i) {
        gcn_wmma_gemm_f32<<<gemm_grid_hid, gemm_blk, 0, stream>>>(
            li, conv_w + (size_t)i * NHID * NHID, nullptr,
            xw, nullptr, NUM_NODES, NHID, NHID);

        gcn_zero_f32<<<elem_blocks, 256, 0, stream>>>(acc, elem_total);

        gcn_spmm_scatter<<<edge_blocks, 256, 0, stream>>>(
            rows, cols, vals, xw, acc, NUM_EDGES);

        gcn_combine_relu<<<elem_blocks, 256, 0, stream>>>(
            acc, h0, conv_b + (size_t)i * NHID, li, elem_total);
    }

    // Stage 3: out = li @ fc1_w + fc1_b
    gcn_wmma_gemm_f32<<<gemm_grid_cls, gemm_blk, 0, stream>>>(
        li, fc1_w, fc1_b, out, nullptr, NUM_NODES, NHID, NCLASS);
}